// SoftCrossEntropyLoss_37864431682699
// MI455X (gfx1250) — compile-verified
//
#include <hip/hip_runtime.h>
#include <hip/hip_bf16.h>

// SoftCrossEntropyLoss, N=1048576 rows, K=256 cols, fp32.
// Memory-bound: 2 GiB streamed once -> ~92 us floor at 23.3 TB/s.
// Layout: 4 rows per wave32, 8 lanes per row, 32 elements per lane.
// Row reductions are 8-lane butterflies done entirely with DPP8 (VALU-only).

constexpr int KCOLS = 256;

typedef float v4f __attribute__((ext_vector_type(4)));
typedef float v2f __attribute__((ext_vector_type(2)));
typedef float v8f __attribute__((ext_vector_type(8)));

// ---- DPP8 xor-butterfly helpers (within groups of 8 lanes) ----
constexpr unsigned dppsel(int a, int b, int c, int d, int e, int f, int g, int h) {
    return (unsigned)a | ((unsigned)b << 3) | ((unsigned)c << 6) | ((unsigned)d << 9) |
           ((unsigned)e << 12) | ((unsigned)f << 15) | ((unsigned)g << 18) | ((unsigned)h << 21);
}
constexpr unsigned SEL_X1 = dppsel(1, 0, 3, 2, 5, 4, 7, 6);  // lane ^ 1
constexpr unsigned SEL_X2 = dppsel(2, 3, 0, 1, 6, 7, 4, 5);  // lane ^ 2
constexpr unsigned SEL_X4 = dppsel(4, 5, 6, 7, 0, 1, 2, 3);  // lane ^ 4

template <unsigned SEL>
__device__ __forceinline__ float dpp8_movf(float x) {
    return __int_as_float(__builtin_amdgcn_mov_dpp8(__float_as_int(x), SEL));
}
__device__ __forceinline__ float group8_sum(float v) {
    v += dpp8_movf<SEL_X4>(v);
    v += dpp8_movf<SEL_X2>(v);
    v += dpp8_movf<SEL_X1>(v);
    return v;
}
__device__ __forceinline__ float group8_max(float v) {
    v = fmaxf(v, dpp8_movf<SEL_X4>(v));
    v = fmaxf(v, dpp8_movf<SEL_X2>(v));
    v = fmaxf(v, dpp8_movf<SEL_X1>(v));
    return v;
}

// ds_swizzle xor butterfly (group-of-32 mode: and=0x1f, or=0, xor=M) — used
// only once per wave at loop end, so its DScnt wait is negligible.
template <int M>
__device__ __forceinline__ float swz_xor_add(float v) {
    return v + __int_as_float(
        __builtin_amdgcn_ds_swizzle(__float_as_int(v), 0x1f | (M << 10)));
}

__device__ __forceinline__ v4f vmax4(v4f a, v4f b) {
    v4f r;
    r.x = fmaxf(a.x, b.x); r.y = fmaxf(a.y, b.y);
    r.z = fmaxf(a.z, b.z); r.w = fmaxf(a.w, b.w);
    return r;
}

__global__ void sce_rows_kernel(const float* __restrict__ x,
                                const float* __restrict__ t,
                                float* __restrict__ partials,
                                int nrows) {   // nrows assumed multiple of 4
    __shared__ float wave_part[8];  // 256 threads = 8 waves
    const int lane        = threadIdx.x & 31;
    const int sub         = lane & 7;        // lane within row group
    const int grp         = lane >> 3;       // which of 4 rows this wave
    const int waveInBlock = threadIdx.x >> 5;
    const int wave        = (int)((blockIdx.x * blockDim.x + threadIdx.x) >> 5);
    const int totalWaves  = (int)((gridDim.x * blockDim.x) >> 5);

    float acc = 0.0f;

    for (int base = wave * 4; base < nrows; base += totalWaves * 4) {
        const int row = base + grp;
        const v4f* xv = reinterpret_cast<const v4f*>(x) + (size_t)row * (KCOLS / 4);
        const v4f* tv = reinterpret_cast<const v4f*>(t) + (size_t)row * (KCOLS / 4);

        // Each lane: 8 b128 loads of x, kept in registers for the exp pass.
        v4f xr[8];
#pragma unroll
        for (int j = 0; j < 8; ++j)
            xr[j] = __builtin_nontemporal_load(xv + sub + 8 * j);

        // Row max (lane-local tree, then 3-step DPP8 butterfly over 8 lanes).
        v4f m4 = vmax4(vmax4(xr[0], xr[1]), vmax4(xr[2], xr[3]));
        m4 = vmax4(m4, vmax4(vmax4(xr[4], xr[5]), vmax4(xr[6], xr[7])));
        float m = fmaxf(fmaxf(m4.x, m4.y), fmaxf(m4.z, m4.w));
        m = group8_max(m);

        // sum exp(x - m)
        float se = 0.0f;
#pragma unroll
        for (int j = 0; j < 8; ++j) {
            se += __expf(xr[j].x - m) + __expf(xr[j].y - m)
                + __expf(xr[j].z - m) + __expf(xr[j].w - m);
        }

        // dot(t, x) and sum(t), consuming t on the fly.
        v4f dot4 = {0.f, 0.f, 0.f, 0.f};
        v4f st4  = {0.f, 0.f, 0.f, 0.f};
#pragma unroll
        for (int j = 0; j < 8; ++j) {
            v4f tt = __builtin_nontemporal_load(tv + sub + 8 * j);
            dot4 += tt * xr[j];
            st4  += tt;
        }
        float dot = (dot4.x + dot4.y) + (dot4.z + dot4.w);
        float st  = (st4.x + st4.y) + (st4.z + st4.w);

        se  = group8_sum(se);
        dot = group8_sum(dot);
        st  = group8_sum(st);

        float lse = m + __logf(se);
        acc += lse * st - dot;   // identical across the 8 lanes of each group
    }

    // Combine the 4 row-groups of this wave (once per kernel, not per row).
    // Lane L adds one representative of each other group: after xor-8 and
    // xor-16 every lane holds G0+G1+G2+G3 exactly once (no replica scaling).
    acc = swz_xor_add<8>(acc);
    acc = swz_xor_add<16>(acc);

    if (lane == 0) wave_part[waveInBlock] = acc;
    __syncthreads();
    if (threadIdx.x == 0) {
        float s = 0.0f;
#pragma unroll
        for (int i = 0; i < 8; ++i) s += wave_part[i];
        partials[blockIdx.x] = s;
    }
}

// Single-wave final reduction. V_WMMA_F32_16X16X4_F32 with all-ones B, zero C:
// sum(all 256 D entries) = 16 * sum(all A entries), independent of lane striping.
__global__ void sce_final_kernel(const float* __restrict__ partials,
                                 int nparts,
                                 float* __restrict__ out,
                                 float invN) {
    const int lane = threadIdx.x;  // launched with exactly 32 threads
    float v = 0.0f;
    for (int i = lane; i < nparts; i += 32) v += partials[i];

    v2f a; a.x = v;    a.y = 0.0f;
    v2f b; b.x = 1.0f; b.y = 1.0f;
    v8f c = {};
    v8f d = __builtin_amdgcn_wmma_f32_16x16x4_f32(
        /*neg_a=*/false, a, /*neg_b=*/false, b,
        /*c_mod=*/(short)0, c, /*reuse_a=*/false, /*reuse_b=*/false);

    float q = d[0] + d[1] + d[2] + d[3] + d[4] + d[5] + d[6] + d[7];
    // wave-wide butterfly: all lanes end with 16 * total
    q = swz_xor_add<16>(q);
    q = swz_xor_add<8>(q);
    q = swz_xor_add<4>(q);
    q = swz_xor_add<2>(q);
    q = swz_xor_add<1>(q);

    if (lane == 0) out[0] = q * (1.0f / 16.0f) * invN;
}

extern "C" void kernel_launch(void* const* d_in, const int* in_sizes, int n_in,
                              void* d_out, int out_size, void* d_ws, size_t ws_size,
                              hipStream_t stream) {
    const float* x = (const float*)d_in[0];
    const float* t = (const float*)d_in[1];
    float* out = (float*)d_out;
    float* partials = (float*)d_ws;

    const int nrows = in_sizes[0] / KCOLS;  // 1048576

    int blocks = 4096;                      // 32768 waves, 32 rows/wave
    size_t need = (size_t)blocks * sizeof(float);
    if (need > ws_size) blocks = (int)(ws_size / sizeof(float));
    if (blocks < 1) blocks = 1;

    sce_rows_kernel<<<blocks, 256, 0, stream>>>(x, t, partials, nrows);
    sce_final_kernel<<<1, 32, 0, stream>>>(partials, blocks, out, 1.0f / (float)nrows);
}